// GeometricStrideUnpool_51573967290684
// MI455X (gfx1250) — compile-verified
//
#include <hip/hip_runtime.h>
#include <stdint.h>

typedef __attribute__((ext_vector_type(2))) float v2f;
typedef __attribute__((ext_vector_type(8))) float v8f;

#define DDIM   512
#define LDIM   2048
#define BDIM   4
#define KFAC   4
#define NCHUNK 6
#define KB     64
#define MT     128
#define NT     256
#define GS     68   /* gbuf row stride (floats), ==4 mod 64 -> conflict-free b64 reads */
#define BS     68   /* blds per-column stride (floats) */
#define XS     516  /* xbuf row stride */
#define WS     516  /* wbuf row stride */
#define GBUF_F (MT * GS)
#define BLDS_F (NT * BS)
#define XBUF_F (33 * XS)
#define WBUF_F (KFAC * WS)
#define LDS_FLOATS (GBUF_F + BLDS_F + XBUF_F + WBUF_F)

__device__ __forceinline__ void async_copy_b128(uint32_t lds_byte_off, uint64_t gaddr) {
  // CDNA5 async global->LDS copy, 16B per lane, tracked by ASYNCcnt
  asm volatile("global_load_async_to_lds_b128 %0, %1, off"
               :: "v"(lds_byte_off), "v"(gaddr)
               : "memory");
}

__global__ __launch_bounds__(512, 1)
void geo_unpool_fused(const float* __restrict__ x,
                      const float* __restrict__ W,
                      const float* __restrict__ pk,
                      const float* __restrict__ bias,
                      float* __restrict__ out) {
  extern __shared__ float lds[];
  float* gbuf = lds;                              // g tile   [128][GS]
  float* blds = lds + GBUF_F;                     // B tile   [256][BS] (k-contiguous per n)
  float* xbuf = lds + GBUF_F + BLDS_F;            // x rows   [33][XS]  (row t = x[l0-1+t])
  float* wbuf = lds + GBUF_F + BLDS_F + XBUF_F;   // W rows   [4][WS]

  const int tid = threadIdx.x;
  const int wgn = blockIdx.x;            // N half: 0..1
  const int wgm = blockIdx.y;            // M stripe: 0..255
  const int m0  = wgm * MT;
  const int bidx = m0 >> 13;             // batch (8192 rows per batch)
  const int l0   = (m0 & 8191) >> 2;     // first l of stripe (32 l values)
  const int n0g  = wgn * NT;

  // ---- stage x rows [l0-1 .. l0+31] and W into LDS via async copies ----
  {
    const int rlo = (l0 == 0) ? 1 : 0;   // row 0 must be zeros when l0==0
    const float* xrow = x + ((int64_t)(bidx * LDIM + (l0 - 1)) * DDIM);
    const int first = rlo * (DDIM / 4);
    const int total = 33 * (DDIM / 4);
    for (int idx = first + tid; idx < total; idx += 512) {
      const int row = idx >> 7;
      const int off = (idx & 127) << 2;
      async_copy_b128((uint32_t)(uintptr_t)&xbuf[row * XS + off],
                      (uint64_t)(uintptr_t)(xrow + (int64_t)row * DDIM + off));
    }
    {
      const int row = tid >> 7;          // 4 rows * 128 chunks = 512 threads
      const int off = (tid & 127) << 2;
      async_copy_b128((uint32_t)(uintptr_t)&wbuf[row * WS + off],
                      (uint64_t)(uintptr_t)(W + row * DDIM + off));
    }
    if (l0 == 0 && tid < DDIM / 4) {
      float4 z = make_float4(0.f, 0.f, 0.f, 0.f);
      *(float4*)&xbuf[tid << 2] = z;     // coarse_prev at l=0 is zero padding
    }
    asm volatile("s_wait_asynccnt 0" ::: "memory");
  }
  __syncthreads();

  v8f acc[8] = {};

  // wave/lane mapping for the GEMM phase: 16 waves = 8 row-blocks x 2 col-halves
  const int lane = tid & 31;
  const int wv   = tid >> 5;
  const int rb   = wv & 7;               // 16-row block within the 128-row stripe
  const int ch   = wv >> 3;              // 128-col half within the 256-col tile
  const int lm   = lane & 15;
  const int koff = (lane >> 4) << 1;     // A/B fragment K offset: lanes16-31 -> K+2

  // thread mapping for g generation: row = tid&127, 16 consecutive columns
  const int gr   = tid & 127;
  const int iseg = tid >> 7;
  const int gj   = gr & 3;               // k-index j (selects W row)
  const int lrow = gr >> 2;              // l - l0
  const int srow = (gj < 3) ? lrow : (lrow + 1);  // j<3 -> coarse_prev, j==3 -> x[l]
  const float* xsrc = &xbuf[srow * XS];
  const float* wjp  = &wbuf[gj * WS];
  float* gdst = &gbuf[gr * GS + iseg * 16];

  // thread mapping for B staging (transpose to k-contiguous layout)
  const int bn  = tid & 255;
  const int bk0 = (tid >> 8) << 5;       // 0 or 32
  float* bdst = &blds[bn * BS + bk0];

  const float* ga  = &gbuf[(rb * 16 + lm) * GS + koff];
  const float* bbp = &blds[(ch * 128 + lm) * BS + koff];

  for (int q = 0; q < NCHUNK; ++q) {
    const int s = 1 << (q >> 1);         // SHIFTS = 1,1,2,2,4,4
    const bool even = (q & 1) == 0;      // even chunk: sw - W*src_s ; odd: silu(sw)
    for (int kb = 0; kb < DDIM / KB; ++kb) {
      // ---- phase A: build 128x64 g tile + stage 64x256 B slab ----
      {
        const int cb = kb * KB + iseg * 16;
        #pragma unroll
        for (int ii = 0; ii < 16; ++ii) {
          const int c  = cb + ii;
          const int cm = (c - s + DDIM) & (DDIM - 1);
          const float srcv = xsrc[c];
          const float srcs = xsrc[cm];
          const float wc   = wjp[c];
          const float ws   = wjp[cm];
          const float sw   = srcv * ws;
          float val;
          if (even) {
            val = __builtin_fmaf(-wc, srcs, sw);
          } else {
            val = sw / (1.0f + __expf(-sw));   // silu
          }
          gdst[ii] = val;
        }
        const float* bsrc = pk + (size_t)(q * DDIM + kb * KB + bk0) * DDIM + (n0g + bn);
        #pragma unroll 8
        for (int kk2 = 0; kk2 < 32; ++kk2) {
          bdst[kk2] = bsrc[(size_t)kk2 * DDIM];  // coalesced global, transposed to LDS
        }
      }
      __syncthreads();
      // ---- phase B: fp32 WMMA over this 64-wide K block ----
      #pragma unroll 4
      for (int kk = 0; kk < KB; kk += 4) {
        const v2f a = *(const v2f*)(ga + kk);                  // ds_load_b64, conflict-free
        #pragma unroll
        for (int t = 0; t < 8; ++t) {
          const v2f bv = *(const v2f*)(bbp + t * 16 * BS + kk);
          acc[t] = __builtin_amdgcn_wmma_f32_16x16x4_f32(
              false, a, false, bv, (short)0, acc[t], false, false);
        }
      }
      __syncthreads();
    }
  }

  // ---- epilogue: add bias, stream out with nontemporal stores ----
  const int mb = m0 + rb * 16 + ((lane >> 4) << 3);   // lanes16-31 hold M+8 rows
  #pragma unroll
  for (int t = 0; t < 8; ++t) {
    const int n = n0g + ch * 128 + t * 16 + lm;
    const float bsv = bias[n];
    #pragma unroll
    for (int v = 0; v < 8; ++v) {
      const float val = acc[t][v] + bsv;
      __builtin_nontemporal_store(val, &out[(size_t)(mb + v) * DDIM + n]);
    }
  }
}

extern "C" void kernel_launch(void* const* d_in, const int* in_sizes, int n_in,
                              void* d_out, int out_size, void* d_ws, size_t ws_size,
                              hipStream_t stream) {
  const float* x    = (const float*)d_in[0];
  const float* W    = (const float*)d_in[1];
  const float* pk   = (const float*)d_in[2];
  const float* bias = (const float*)d_in[3];
  float* out = (float*)d_out;

  const size_t shmem = (size_t)LDS_FLOATS * sizeof(float);  // ~177 KB dynamic LDS
  (void)hipFuncSetAttribute((const void*)geo_unpool_fused,
                            hipFuncAttributeMaxDynamicSharedMemorySize, (int)shmem);

  dim3 grid(DDIM / NT, (BDIM * LDIM * KFAC) / MT, 1);   // (2, 256)
  dim3 block(512, 1, 1);                                 // 16 wave32 waves
  geo_unpool_fused<<<grid, block, shmem, stream>>>(x, W, pk, bias, out);
}